// CPRL_24292335026358
// MI455X (gfx1250) — compile-verified
//
#include <hip/hip_runtime.h>

// CP regression layer, factorized (no full W tensor):
//   h[(b,c),r] = f1[b,r]*f2[c,r]
//   u[m,r]     = sum_k X[m,k]*h[k,r]          (m=(x,a), k=(b,c))  <-- WMMA f32 GEMM
//   t[x,r]     = sum_a f0[a,r]*u[(x,a),r]
//   out[x,d]   = bias + sum_r w[r]*f3[d,r]*t[x,r]

typedef float v2f __attribute__((ext_vector_type(2)));
typedef float v8f __attribute__((ext_vector_type(8)));

#define B_   32
#define D1_  128
#define D2_  64
#define D3_  32
#define O1_  64
#define R_   16
#define M_   (B_ * D1_)       // 4096 rows
#define K_   (D2_ * D3_)      // 2048 reduction
#define KCHUNKS 4
#define KC   (K_ / KCHUNKS)   // 512 k per chunk
#define TILES_PER_WAVE 2
#define WAVES_PER_BLOCK 4
#define TILES_PER_BLOCK (TILES_PER_WAVE * WAVES_PER_BLOCK)  // 8
#define MTILES (M_ / 16)      // 256
#define GRID_X (MTILES / TILES_PER_BLOCK)                   // 32

// ---------------------------------------------------------------------------
// Kernel 1: h[(b*32+c)*16 + r] = f1[b,r] * f2[c,r]   (2048 x 16 fp32, 128KB)
// ---------------------------------------------------------------------------
__global__ void cprl_build_h(const float* __restrict__ f1,
                             const float* __restrict__ f2,
                             float* __restrict__ h) {
    int i = blockIdx.x * blockDim.x + threadIdx.x;   // 0 .. 32767
    if (i >= D2_ * D3_ * R_) return;
    int r  = i & (R_ - 1);
    int bc = i >> 4;
    int c  = bc & (D3_ - 1);
    int b  = bc >> 5;
    h[i] = f1[b * R_ + r] * f2[c * R_ + r];
}

// ---------------------------------------------------------------------------
// Kernel 2: u partials via V_WMMA_F32_16X16X4_F32.
//   A-frag (16x4 f32): lane<16 holds (K0,K1) of row lane; lane>=16 holds (K2,K3)
//   B-frag (4x16 f32): mirrored layout, read from LDS copy of h chunk
//   D (16x16 f32, 8 VGPRs): M = v + 8*(lane>>4), N = lane&15
// ---------------------------------------------------------------------------
__global__ void __launch_bounds__(128)
cprl_gemm(const float* __restrict__ X,   // (4096, 2048) row-major
          const float* __restrict__ h,   // (2048, 16)
          float* __restrict__ pu) {      // (KCHUNKS, 4096, 16)
    __shared__ float hs[KC * R_];        // 512*16 fp32 = 32 KB of the 320 KB WGP LDS

    const int tid   = threadIdx.x;
    const int chunk = blockIdx.y;
    const int k0    = chunk * KC;

    // Cooperative 32KB copy of this K-chunk of h into LDS (contiguous, b128 loads).
    {
        const float4* src = (const float4*)(h + (size_t)k0 * R_);
        float4*       dst = (float4*)hs;
        for (int i = tid; i < (KC * R_) / 4; i += 128) dst[i] = src[i];
    }
    __syncthreads();

    const int wave  = tid >> 5;
    const int lane  = tid & 31;          // wave32
    const int n     = lane & 15;         // N (= r) column
    const int khalf = (lane >> 4) << 1;  // 0 for lanes 0-15, 2 for lanes 16-31

    const int tile0 = blockIdx.x * TILES_PER_BLOCK + wave * TILES_PER_WAVE;

    // Per-tile A base pointer: row = tile*16 + (lane&15), starting K = k0 + khalf
    const float* aBase[TILES_PER_WAVE];
#pragma unroll
    for (int t = 0; t < TILES_PER_WAVE; ++t) {
        int row = (tile0 + t) * 16 + n;
        aBase[t] = X + (size_t)row * K_ + k0 + khalf;
    }

    v8f acc[TILES_PER_WAVE] = {};

    for (int k = 0; k < KC; k += 4) {
        // B fragment: VGPR0 = h[k+khalf, n], VGPR1 = h[k+khalf+1, n]
        v2f b;
        b.x = hs[(k + khalf)     * R_ + n];
        b.y = hs[(k + khalf + 1) * R_ + n];
#pragma unroll
        for (int t = 0; t < TILES_PER_WAVE; ++t) {
            v2f a = *(const v2f*)(aBase[t] + k);   // global_load_b64, 8B/lane
            acc[t] = __builtin_amdgcn_wmma_f32_16x16x4_f32(
                /*neg_a=*/false, a, /*neg_b=*/false, b,
                /*c_mod=*/(short)0, acc[t],
                /*reuse_a=*/false, /*reuse_b=*/false);
        }
    }

    // Scatter D to pu[chunk][m][r] per the 16x16 f32 C/D VGPR layout.
#pragma unroll
    for (int t = 0; t < TILES_PER_WAVE; ++t) {
        int rowb = (tile0 + t) * 16 + ((lane >> 4) << 3);
#pragma unroll
        for (int v = 0; v < 8; ++v) {
            int m = rowb + v;
            pu[((size_t)chunk * M_ + m) * R_ + n] = acc[t][v];
        }
    }
}

// ---------------------------------------------------------------------------
// Kernel 3: out[x,d] = bias + sum_r w[r]*f3[d,r]* sum_a f0[a,r]* sum_ck pu[ck,(x,a),r]
// One block per batch element; deterministic LDS reduction, 16 threads each serially sum 64 partials (no FP atomics).
// ---------------------------------------------------------------------------
__global__ void __launch_bounds__(64)
cprl_finalize(const float* __restrict__ pu,
              const float* __restrict__ f0,
              const float* __restrict__ f3,
              const float* __restrict__ w,
              const float* __restrict__ bias,
              float* __restrict__ out) {
    __shared__ float part[64][R_];
    __shared__ float tsh[R_];
    const int x   = blockIdx.x;
    const int tid = threadIdx.x;

    float loc[R_];
#pragma unroll
    for (int r = 0; r < R_; ++r) loc[r] = 0.0f;

    for (int a = tid; a < D1_; a += 64) {       // 2 'a' values per thread
        int m = x * D1_ + a;
#pragma unroll
        for (int r = 0; r < R_; ++r) {
            float s = 0.0f;
#pragma unroll
            for (int c = 0; c < KCHUNKS; ++c)
                s += pu[((size_t)c * M_ + m) * R_ + r];
            loc[r] += f0[a * R_ + r] * s;
        }
    }
#pragma unroll
    for (int r = 0; r < R_; ++r) part[tid][r] = loc[r];
    __syncthreads();

    if (tid < R_) {
        float s = 0.0f;
        for (int i = 0; i < 64; ++i) s += part[i][tid];
        tsh[tid] = s;
    }
    __syncthreads();

    float o = bias[0];
#pragma unroll
    for (int r = 0; r < R_; ++r) o += w[r] * f3[tid * R_ + r] * tsh[r];
    out[x * O1_ + tid] = o;
}

// ---------------------------------------------------------------------------
extern "C" void kernel_launch(void* const* d_in, const int* in_sizes, int n_in,
                              void* d_out, int out_size, void* d_ws, size_t ws_size,
                              hipStream_t stream) {
    const float* x    = (const float*)d_in[0];
    const float* f0   = (const float*)d_in[1];
    const float* f1   = (const float*)d_in[2];
    const float* f2   = (const float*)d_in[3];
    const float* f3   = (const float*)d_in[4];
    const float* w    = (const float*)d_in[5];
    const float* bias = (const float*)d_in[6];
    float* out = (float*)d_out;

    // Workspace layout (fully overwritten every launch -> no init needed):
    //   h  : K_*R_            floats = 128 KB
    //   pu : KCHUNKS*M_*R_    floats = 1 MB
    float* h  = (float*)d_ws;
    float* pu = h + (size_t)K_ * R_;

    cprl_build_h<<<dim3((K_ * R_) / 256), dim3(256), 0, stream>>>(f1, f2, h);
    cprl_gemm<<<dim3(GRID_X, KCHUNKS), dim3(128), 0, stream>>>(x, h, pu);
    cprl_finalize<<<dim3(B_), dim3(64), 0, stream>>>(pu, f0, f3, w, bias, out);
}